// mfn_8461085573656
// MI455X (gfx1250) — compile-verified
//
#include <hip/hip_runtime.h>
#include <math.h>

// MI455X reasoning:
//  - out = 204.8 MB f32 stores, in = 8.2 MB reads (x2) -> ~9.2us floor @23.3TB/s.
//  - Store-bandwidth bound; arithmetic ~0.15 GFLOP (free).
//  - The 25->625 expansion is expressed as Y = A(16x25) . M(25x625) with M a 0/1
//    shift-selection matrix -> exact in f32 WMMA (v_wmma_f32_16x16x4_f32),
//    so the CDNA5 matrix pipe does the gather at zero accuracy cost.
//  - Output (204.8MB) > L2 (192MB): use nontemporal stores.

typedef __attribute__((ext_vector_type(2))) float v2f;
typedef __attribute__((ext_vector_type(8))) float v8f;

#define BATCH 128
#define CH    640
#define HW    25
#define NEXP  625
#define EPS   1e-12f

// ws layout (floats): s[128*25] | fx[25] | fy[25] | q[25]
#define WS_S  0
#define WS_FX (BATCH * HW)
#define WS_FY (WS_FX + 25)
#define WS_Q  (WS_FY + 25)

// ---------------- Pass 1: per-(b,p) normalize scale + DCT tables --------------
__global__ void mfn_pass1_norms(const float* __restrict__ x, float* __restrict__ ws) {
    const int b   = blockIdx.x;
    const int tid = threadIdx.x;                 // 256 threads
    __shared__ float part[200];

    const int p   = tid >> 3;                    // 8 threads per spatial position
    const int sub = tid & 7;
    if (p < HW) {
        float acc = 0.f;
        const float* xb = x + (size_t)b * CH * HW + p;
        for (int c = sub; c < CH; c += 8) {
            float v = xb[(size_t)c * HW];
            v = v > 0.f ? v : 0.f;               // relu
            acc += v * v;
        }
        part[p * 8 + sub] = acc;
    }
    __syncthreads();

    if (tid < HW) {
        float ssum = 0.f;
#pragma unroll
        for (int i = 0; i < 8; ++i) ssum += part[tid * 8 + i];
        float n1     = sqrtf(ssum);
        float d1     = n1 > EPS ? n1 : EPS;
        float scale1 = 1.f / d1;                 // first normalize
        float n2     = n1 * scale1;              // norm after first normalize
        float d2     = n2 > EPS ? n2 : EPS;
        ws[WS_S + b * HW + tid] = scale1 / d2;   // combined double-normalize scale
    }

    if (blockIdx.x == 0 && tid < 25) {           // DCT tables in f64 (match numpy)
        const double PI = 3.14159265358979323846;
        double t  = (double)tid + 0.5;
        double fx = cos(PI * 75.0 * t / 25.0) / 5.0 * sqrt(2.0);
        double fy = cos(PI * 45.0 * t / 25.0) / 5.0 * sqrt(2.0);
        int i = tid / 5, j = tid % 5;
        int di = i > 2 ? i - 2 : 2 - i;
        int dj = j > 2 ? j - 2 : 2 - j;
        ws[WS_FX + tid] = (float)fx;
        ws[WS_FY + tid] = (float)fy;
        ws[WS_Q  + tid] = 1.0f / (float)(di + dj + 1);
    }
}

// -------- Pass 2: WMMA gather (A . shift-matrix) + diagonal scale + NT store --
// Block: 256 threads (8 waves), handles one batch b and 16 channels.
// Grid:  128 * (640/16) = 5120 blocks.
__global__ void mfn_pass2_wmma(const float* __restrict__ x, const float* __restrict__ ws,
                               float* __restrict__ out) {
    const int blk  = blockIdx.x;
    const int b    = blk / 40;
    const int c0   = (blk % 40) * 16;
    const int tid  = threadIdx.x;
    const int lane = tid & 31;                   // wave32
    const int wave = tid >> 5;                   // 0..7

    __shared__ float xnL[16 * 28];               // A staging, K padded 25->28 (zeros)
    __shared__ float fxL[25], fyL[25], qL[25];

    for (int i = tid; i < 16 * 28; i += 256) xnL[i] = 0.f;
    if (tid < 25) {
        fxL[tid] = ws[WS_FX + tid];
        fyL[tid] = ws[WS_FY + tid];
        qL[tid]  = ws[WS_Q  + tid];
    }
    __syncthreads();

    for (int i = tid; i < 16 * HW; i += 256) {
        int cc = i / HW, p = i % HW;
        float v = x[((size_t)(b * CH + c0 + cc)) * HW + p];
        v = v > 0.f ? v : 0.f;                   // relu
        xnL[cc * 28 + p] = v * ws[WS_S + b * HW + p];
    }
    __syncthreads();

    const int arow = lane & 15;                  // A row (channel) for this lane
    const int hi   = lane >> 4;                  // lane-half selects K/M split

    for (int t = wave; t < 40; t += 8) {         // 40 N-tiles of 16, uniform per wave
        const int n    = t * 16 + (lane & 15);   // output column (p,k) flat, 0..639
        // column -> source spatial position p'' (0/1 selector matrix M)
        const int pcol = n / 25, kcol = n % 25;
        const int hh   = pcol / 5 + kcol / 5 - 2;
        const int ww   = pcol % 5 + kcol % 5 - 2;
        const bool ok  = (n < NEXP) && hh >= 0 && hh < 5 && ww >= 0 && ww < 5;
        const int ptgt = ok ? (hh * 5 + ww) : -1;

        v8f acc = {};                            // C = 0
#pragma unroll
        for (int kk = 0; kk < 7; ++kk) {         // K = 28 in steps of 4
            const int kb = kk * 4 + hi * 2;      // this lane-half's K rows
            v2f a;                               // A 16x4 fragment (from LDS)
            a.x = xnL[arow * 28 + kb];
            a.y = xnL[arow * 28 + kb + 1];
            v2f bsel;                            // B 4x16 fragment: 0/1 selector
            bsel.x = (kb     == ptgt) ? 1.f : 0.f;
            bsel.y = (kb + 1 == ptgt) ? 1.f : 0.f;
            acc = __builtin_amdgcn_wmma_f32_16x16x4_f32(
                false, a, false, bsel, (short)0, acc, false, false);
        }

        if (n < NEXP) {                          // drop pad columns 625..639
            const float wn = qL[kcol] * fxL[pcol] * fyL[kcol];
#pragma unroll
            for (int r = 0; r < 8; ++r) {
                const int cc  = r + hi * 8;      // D row layout: r / r+8 per half
                const float v = acc[r] * xnL[cc * 28 + pcol] * wn;
                __builtin_nontemporal_store(
                    v, &out[((size_t)(b * CH + c0 + cc)) * NEXP + n]);
            }
        }
    }
}

extern "C" void kernel_launch(void* const* d_in, const int* in_sizes, int n_in,
                              void* d_out, int out_size, void* d_ws, size_t ws_size,
                              hipStream_t stream) {
    const float* x   = (const float*)d_in[0];
    float*       out = (float*)d_out;
    float*       ws  = (float*)d_ws;

    mfn_pass1_norms<<<BATCH, 256, 0, stream>>>(x, ws);
    mfn_pass2_wmma<<<BATCH * (CH / 16), 256, 0, stream>>>(x, ws, out);
}